// GridPostProcessor_19937238188294
// MI455X (gfx1250) — compile-verified
//
#include <hip/hip_runtime.h>
#include <math.h>

typedef float v4f __attribute__((ext_vector_type(4)));
typedef int   v4i __attribute__((vector_size(4 * sizeof(int))));   // match builtin param type

#define AS1 __attribute__((address_space(1)))
#define AS3 __attribute__((address_space(3)))

#define GP    9               // GRID_POINTS
#define HALFD 28              // HALF
#define MAPE  (HALFD*HALFD)   // 784 floats per (roi, grid) map = 3136 B
#define VECN  (MAPE/4)        // 196 float4 per map

#if __has_builtin(__builtin_amdgcn_global_load_async_to_lds_b128)
#define USE_ASYNC_LDS 1
#endif

__device__ __forceinline__ void wait_asynccnt0() {
#if __has_builtin(__builtin_amdgcn_s_wait_asynccnt)
    __builtin_amdgcn_s_wait_asynccnt(0);
#else
    asm volatile("s_wait_asynccnt 0x0" ::: "memory");
#endif
}

// One block per ROI. 9 waves (288 threads); wave g reduces map g (max+argmax
// of raw logits -- sigmoid is monotonic so argmax is preserved and the max
// score is sigmoid(max logit)). Thread 0 does the tiny per-ROI epilogue.
__global__ __launch_bounds__(288, 1) void grid_post_kernel(
    const float* __restrict__ logits,
    const float* __restrict__ bboxes,
    float* __restrict__ out, int R)
{
    const int r    = blockIdx.x;
    const int wave = threadIdx.x >> 5;   // 0..8
    const int lane = threadIdx.x & 31;

    __shared__ float s_val[GP];
    __shared__ int   s_idx[GP];

    float best = -INFINITY;
    int   bidx = 0;

#ifdef USE_ASYNC_LDS
    // ---- CDNA5 async data-mover path: stream map into LDS via ASYNCcnt ----
    __shared__ __align__(16) float lbuf[GP * MAPE];   // 28224 B per block

    const float* gbase = logits + (size_t)r * (GP * MAPE) + (size_t)wave * MAPE;
    AS1 v4i* gsrc = (AS1 v4i*)((AS1 char*)(char*)gbase + lane * 16);
    AS3 v4i* ldst = (AS3 v4i*)((AS3 char*)(char*)(lbuf + wave * MAPE) + lane * 16);

    // 6 full-wave b128 async copies: cover float4 indices lane + 32k, k=0..5
    __builtin_amdgcn_global_load_async_to_lds_b128(gsrc, ldst, 0,    0);
    __builtin_amdgcn_global_load_async_to_lds_b128(gsrc, ldst, 512,  0);
    __builtin_amdgcn_global_load_async_to_lds_b128(gsrc, ldst, 1024, 0);
    __builtin_amdgcn_global_load_async_to_lds_b128(gsrc, ldst, 1536, 0);
    __builtin_amdgcn_global_load_async_to_lds_b128(gsrc, ldst, 2048, 0);
    __builtin_amdgcn_global_load_async_to_lds_b128(gsrc, ldst, 2560, 0);
    // tail: float4 indices 192..195 -> only lanes 0..3 (EXEC-masked per lane;
    // also avoids reading past the end of the buffer on the final map)
    if (lane < (VECN - 192)) {
        __builtin_amdgcn_global_load_async_to_lds_b128(gsrc, ldst, 3072, 0);
    }
    wait_asynccnt0();

    const v4f* base = (const v4f*)(lbuf + wave * MAPE);   // LDS reads below
#else
    // ---- fallback: direct NT-hinted b128 global loads ----
    const v4f* base = (const v4f*)logits + (size_t)r * (GP * VECN)
                                         + (size_t)wave * VECN;
#endif

    // Indices increase within a lane, so strict '>' keeps first occurrence.
#pragma unroll
    for (int k = 0; k < 6; ++k) {
        const int v = lane + 32 * k;
#ifdef USE_ASYNC_LDS
        const v4f d = base[v];
#else
        const v4f d = __builtin_nontemporal_load((v4f*)(base + v));
#endif
        const int i0 = v * 4;
        if (d.x > best) { best = d.x; bidx = i0;     }
        if (d.y > best) { best = d.y; bidx = i0 + 1; }
        if (d.z > best) { best = d.z; bidx = i0 + 2; }
        if (d.w > best) { best = d.w; bidx = i0 + 3; }
    }
    if (lane < (VECN - 192)) {                 // tail float4s 192..195
        const int v = lane + 192;
#ifdef USE_ASYNC_LDS
        const v4f d = base[v];
#else
        const v4f d = __builtin_nontemporal_load((v4f*)(base + v));
#endif
        const int i0 = v * 4;
        if (d.x > best) { best = d.x; bidx = i0;     }
        if (d.y > best) { best = d.y; bidx = i0 + 1; }
        if (d.z > best) { best = d.z; bidx = i0 + 2; }
        if (d.w > best) { best = d.w; bidx = i0 + 3; }
    }

    // Wave32 butterfly reduce (max, first-index tie-break).
#pragma unroll
    for (int off = 16; off > 0; off >>= 1) {
        const float ov = __shfl_xor(best, off, 32);
        const int   oi = __shfl_xor(bidx, off, 32);
        if (ov > best || (ov == best && oi < bidx)) { best = ov; bidx = oi; }
    }

    if (lane == 0) { s_val[wave] = best; s_idx[wave] = bidx; }
    __syncthreads();

    if (threadIdx.x == 0) {
        const float b0 = bboxes[r*4+0], b1 = bboxes[r*4+1];
        const float b2 = bboxes[r*4+2], b3 = bboxes[r*4+3];
        const float w   = b2 - b0,      h   = b3 - b1;
        const float x1m = b0 - 0.5f*w,  y1m = b1 - 0.5f*h;

        // subs offsets for GRID_SIZE=3, WHOLE_MAP=56, HALF=28: {0,14,28}
        const int subs[3] = {0, 14, 28};

        float ax[GP], ay[GP], sc[GP];
#pragma unroll
        for (int g = 0; g < GP; ++g) {
            const int pos = s_idx[g];
            const int xs  = (pos % HALFD) + subs[g % 3];
            const int ys  = (pos / HALFD) + subs[g / 3];
            // (xs+0.5)/(2*28) * (1+1) * w + x1m  ==  (xs+0.5)/28 * w + x1m
            ax[g] = ((float)xs + 0.5f) * (1.0f / 28.0f) * w + x1m;
            ay[g] = ((float)ys + 0.5f) * (1.0f / 28.0f) * h + y1m;
            sc[g] = 1.0f / (1.0f + expf(-s_val[g]));   // sigmoid(max logit)
        }

        v4f o;
        o.x = (ax[0]*sc[0] + ax[1]*sc[1] + ax[2]*sc[2]) / (sc[0] + sc[1] + sc[2]); // x1: 0,1,2
        o.y = (ay[0]*sc[0] + ay[3]*sc[3] + ay[6]*sc[6]) / (sc[0] + sc[3] + sc[6]); // y1: 0,3,6
        o.z = (ax[6]*sc[6] + ax[7]*sc[7] + ax[8]*sc[8]) / (sc[6] + sc[7] + sc[8]); // x2: 6,7,8
        o.w = (ay[2]*sc[2] + ay[5]*sc[5] + ay[8]*sc[8]) / (sc[2] + sc[5] + sc[8]); // y2: 2,5,8
        *(v4f*)(out + (size_t)r * 4) = o;
    }
}

extern "C" void kernel_launch(void* const* d_in, const int* in_sizes, int n_in,
                              void* d_out, int out_size, void* d_ws, size_t ws_size,
                              hipStream_t stream) {
    const float* logits = (const float*)d_in[0];
    const float* bboxes = (const float*)d_in[1];
    float*       out    = (float*)d_out;
    const int R = in_sizes[0] / (GP * MAPE);   // 50000 for the reference setup
    grid_post_kernel<<<dim3(R), dim3(288), 0, stream>>>(logits, bboxes, out, R);
}